// SEFusion_45921790329192
// MI455X (gfx1250) — compile-verified
//
#include <hip/hip_runtime.h>
#include <math.h>

typedef __attribute__((ext_vector_type(2))) float v2f;
typedef __attribute__((ext_vector_type(8))) float v8f;

#define NTOK 25
#define BS 8
#define MAXC 5
#define CC 256
#define HH 48
#define WW 176
#define CHW (CC*HH*WW)        // 2,162,688 elements per token
#define CHW4 (CHW/4)          // 540,672 float4 per token
#define BLKS_PER_TOK 132
#define VEC_PER_THREAD 16     // 132 * 256 * 16 == 540,672
#define BLKS_PER_B 2112       // 540,672 / 256

// ---------------- Phase 1: per-token partial sum/max (float4 loads) ----------------
__global__ void partial_reduce_kernel(const float* __restrict__ x,
                                      float* __restrict__ psum,
                                      float* __restrict__ pmax) {
    const int tok = blockIdx.x / BLKS_PER_TOK;
    const int blk = blockIdx.x % BLKS_PER_TOK;
    const float4* xv = (const float4*)x;
    const long base = (long)tok * CHW4 + (long)blk * (256 * VEC_PER_THREAD);

    float s = 0.0f;
    float m = -INFINITY;
#pragma unroll
    for (int i = 0; i < VEC_PER_THREAD; ++i) {
        float4 v = xv[base + (long)i * 256 + threadIdx.x];
        s += (v.x + v.y) + (v.z + v.w);
        m = fmaxf(m, fmaxf(fmaxf(v.x, v.y), fmaxf(v.z, v.w)));
    }

    __shared__ float ss[256];
    __shared__ float sm[256];
    ss[threadIdx.x] = s;
    sm[threadIdx.x] = m;
    __syncthreads();
    for (int off = 128; off > 0; off >>= 1) {
        if (threadIdx.x < off) {
            ss[threadIdx.x] += ss[threadIdx.x + off];
            sm[threadIdx.x] = fmaxf(sm[threadIdx.x], sm[threadIdx.x + off]);
        }
        __syncthreads();
    }
    if (threadIdx.x == 0) {
        psum[blockIdx.x] = ss[0];
        pmax[blockIdx.x] = sm[0];
    }
}

// ---------------- Phase 2: deterministic final per-token reduction ----------------
__global__ void final_reduce_kernel(const float* __restrict__ psum,
                                    const float* __restrict__ pmax,
                                    float* __restrict__ tsum,
                                    float* __restrict__ tmax) {
    const int tok = blockIdx.x;
    const int t = threadIdx.x;
    float s = (t < BLKS_PER_TOK) ? psum[tok * BLKS_PER_TOK + t] : 0.0f;
    float m = (t < BLKS_PER_TOK) ? pmax[tok * BLKS_PER_TOK + t] : -INFINITY;

    __shared__ float ss[256];
    __shared__ float sm[256];
    ss[t] = s;
    sm[t] = m;
    __syncthreads();
    for (int off = 128; off > 0; off >>= 1) {
        if (t < off) {
            ss[t] += ss[t + off];
            sm[t] = fmaxf(sm[t], sm[t + off]);
        }
        __syncthreads();
    }
    if (t == 0) {
        tsum[tok] = ss[0];
        tmax[tok] = sm[0];
    }
}

// ---------------- Phase 3: gate MLP on one wave32 via V_WMMA_F32_16X16X4_F32 ----------------
// D = relu(sigmoid(Xsq @ W1T) @ W2T); wEff[b][m] = D[b][m] * conv_w[m]
__global__ void gate_kernel(const int* __restrict__ rec,
                            const float* __restrict__ tsum,
                            const float* __restrict__ tmax,
                            const float* __restrict__ W1,
                            const float* __restrict__ W2,
                            const float* __restrict__ convw,
                            float* __restrict__ wEff) {
    __shared__ float Xsq[16][12];   // [b][k], k = 0..9 valid (avg | max), padded to 12
    __shared__ float B1[12][16];    // W1T: [k][n]
    __shared__ float B2[8][16];     // W2T: [k][n], K padded 5 -> 8
    __shared__ float Sg[16][16];    // sigmoid(stage-1) staging for A-layout reload
    __shared__ int starts[8];

    const int lane = threadIdx.x;

    for (int i = lane; i < 16 * 12; i += 32) (&Xsq[0][0])[i] = 0.0f;
    for (int i = lane; i < 12 * 16; i += 32) (&B1[0][0])[i] = 0.0f;
    for (int i = lane; i < 8 * 16;  i += 32) (&B2[0][0])[i] = 0.0f;
    if (lane == 0) {
        int s = 0;
        for (int j = 0; j < BS; ++j) { starts[j] = s; s += rec[j]; }
    }
    __syncthreads();

    // fill x_sq = [avg | max]; padded slots stay 0 (mean/max of zero padding)
    for (int i = lane; i < BS * MAXC; i += 32) {
        const int b = i / MAXC, m = i % MAXC;
        if (m < rec[b]) {
            const int t = starts[b] + m;
            Xsq[b][m]        = tsum[t] * (1.0f / (float)CHW);
            Xsq[b][MAXC + m] = tmax[t];
        }
    }
    for (int i = lane; i < 5 * 10; i += 32) { const int n = i / 10, k = i % 10; B1[k][n] = W1[i]; }
    for (int i = lane; i < 5 * 5;  i += 32) { const int n = i / 5,  k = i % 5;  B2[k][n] = W2[i]; }
    __syncthreads();

    const int hi = lane >> 4;   // lane group
    const int lo = lane & 15;

    // Stage 1: S = Xsq(16x12) @ W1T(12x16), three K=4 chunks
    v8f c = {};
#pragma unroll
    for (int q = 0; q < 3; ++q) {
        v2f a, b;
        a.x = Xsq[lo][4 * q + 2 * hi + 0];
        a.y = Xsq[lo][4 * q + 2 * hi + 1];
        b.x = B1[4 * q + 2 * hi + 0][lo];
        b.y = B1[4 * q + 2 * hi + 1][lo];
        c = __builtin_amdgcn_wmma_f32_16x16x4_f32(false, a, false, b, (short)0, c,
                                                  false, false);
    }

    // sigmoid, stage to LDS (C/D layout -> A layout needs cross-lane move)
#pragma unroll
    for (int v = 0; v < 8; ++v) Sg[v + 8 * hi][lo] = 1.0f / (1.0f + __expf(-c[v]));
    __syncthreads();

    // Stage 2: D = Sg(16x16, K valid 0..4) @ W2T(8x16), two K=4 chunks (rows >=5 of B2 are zero)
    v8f c2 = {};
#pragma unroll
    for (int q = 0; q < 2; ++q) {
        v2f a, b;
        a.x = Sg[lo][4 * q + 2 * hi + 0];
        a.y = Sg[lo][4 * q + 2 * hi + 1];
        b.x = B2[4 * q + 2 * hi + 0][lo];
        b.y = B2[4 * q + 2 * hi + 1][lo];
        c2 = __builtin_amdgcn_wmma_f32_16x16x4_f32(false, a, false, b, (short)0, c2,
                                                   false, false);
    }

    // wEff[b][m] = relu(D[b][m]) * conv_w[m]; rows b=0..7 live in v=0..7 of lanes hi==0
    if (hi == 0 && lo < MAXC) {
        const float cw = convw[lo];
#pragma unroll
        for (int v = 0; v < 8; ++v)
            wEff[v * MAXC + lo] = fmaxf(c2[v], 0.0f) * cw;
    }
}

// ---------------- Phase 4: fused weighted-sum + bias + relu (float4) ----------------
__global__ void fuse_kernel(const float* __restrict__ x,
                            const int* __restrict__ rec,
                            const float* __restrict__ wEff,
                            const float* __restrict__ convb,
                            float* __restrict__ out) {
    const int b = blockIdx.x / BLKS_PER_B;               // whole block shares one batch
    const long inner = (long)(blockIdx.x % BLKS_PER_B) * 256 + threadIdx.x;

    int start = 0, r = 0;
#pragma unroll
    for (int j = 0; j < BS; ++j) {
        const int rl = rec[j];
        if (j < b) start += rl;
        if (j == b) r = rl;
    }

    const float4* xv = (const float4*)x;
    const float bias = convb[0];
    float ax = bias, ay = bias, az = bias, aw = bias;

    for (int m = 0; m < r; ++m) {
        const float w = wEff[b * MAXC + m];
        float4 v = xv[(long)(start + m) * CHW4 + inner];
        ax = fmaf(w, v.x, ax);
        ay = fmaf(w, v.y, ay);
        az = fmaf(w, v.z, az);
        aw = fmaf(w, v.w, aw);
    }

    float4 o;
    o.x = fmaxf(ax, 0.0f);
    o.y = fmaxf(ay, 0.0f);
    o.z = fmaxf(az, 0.0f);
    o.w = fmaxf(aw, 0.0f);
    ((float4*)out)[(long)b * CHW4 + inner] = o;
}

extern "C" void kernel_launch(void* const* d_in, const int* in_sizes, int n_in,
                              void* d_out, int out_size, void* d_ws, size_t ws_size,
                              hipStream_t stream) {
    const float* x     = (const float*)d_in[0];
    const int*   rec   = (const int*)d_in[1];
    const float* W1    = (const float*)d_in[2];
    const float* W2    = (const float*)d_in[3];
    const float* convw = (const float*)d_in[4];
    const float* convb = (const float*)d_in[5];
    float* out = (float*)d_out;

    // workspace layout (floats): psum[3300] | pmax[3300] | tsum[25] | tmax[25] | wEff[40]
    float* ws   = (float*)d_ws;
    float* psum = ws;
    float* pmax = ws + NTOK * BLKS_PER_TOK;        // +3300
    float* tsum = ws + 2 * NTOK * BLKS_PER_TOK;    // +6600
    float* tmax = tsum + 32;                       // +6632
    float* wEff = tsum + 64;                       // +6664

    partial_reduce_kernel<<<NTOK * BLKS_PER_TOK, 256, 0, stream>>>(x, psum, pmax);
    final_reduce_kernel<<<NTOK, 256, 0, stream>>>(psum, pmax, tsum, tmax);
    gate_kernel<<<1, 32, 0, stream>>>(rec, tsum, tmax, W1, W2, convw, wEff);
    fuse_kernel<<<BS * BLKS_PER_B, 256, 0, stream>>>(x, rec, wEff, convb, out);
}